// TripleScatterModule_71476845740842
// MI455X (gfx1250) — compile-verified
//
#include <hip/hip_runtime.h>
#include <hip/hip_bf16.h>

typedef __attribute__((ext_vector_type(2))) float v2f;
typedef __attribute__((ext_vector_type(8))) float v8f;

#define F_IN   16
#define HID    64
#define F_OUT  16
#define R_DIM  256
#define C_DIM  4096
#define NSETS  2
#define L_DIM  4096
#define K3F    48              // 3 * F_IN
#define RC     (R_DIM * C_DIM) // 1048576

// ---------------------------------------------------------------------------
// Preprocessing: inverse permutation + fused gather/scatter column tables
// ---------------------------------------------------------------------------
__global__ void build_inv_kernel(const int* __restrict__ i0,
                                 const int* __restrict__ i1,
                                 const int* __restrict__ i2,
                                 int* __restrict__ inv) {
  int tid = blockIdx.x * blockDim.x + threadIdx.x;
  if (tid >= NSETS * 3 * L_DIM) return;
  int j = tid & (L_DIM - 1);
  int k = (tid >> 12) % 3;
  int s = tid / (3 * L_DIM);
  const int* ind = (k == 0 ? i0 : (k == 1 ? i1 : i2)) + s * (2 * L_DIM);
  inv[((s * 3 + k) << 12) + ind[2 * j]] = j;   // ind[:,0] is a bijection
}

__global__ void build_srcdst_kernel(const int* __restrict__ i0,
                                    const int* __restrict__ i1,
                                    const int* __restrict__ i2,
                                    const int* __restrict__ mix,
                                    const int* __restrict__ inv,
                                    int* __restrict__ src,
                                    int* __restrict__ dst) {
  int tid = blockIdx.x * blockDim.x + threadIdx.x;   // == (s*3+k)*L + m
  if (tid >= NSETS * 3 * L_DIM) return;
  int k = (tid >> 12) % 3;
  int s = tid / (3 * L_DIM);
  const int* ind = (k == 0 ? i0 : (k == 1 ? i1 : i2)) + s * (2 * L_DIM);
  int mi = mix[tid];
  dst[tid] = ind[2 * mi + 1];                                  // scatter column
  src[tid] = ind[2 * inv[((s * 3 + k) << 12) + mi] + 1];       // gather column
}

__global__ void zero_kernel(float4* __restrict__ p, int n4) {
  int i = blockIdx.x * blockDim.x + threadIdx.x;
  if (i < n4) p[i] = make_float4(0.f, 0.f, 0.f, 0.f);
}

// x[f][r][c]  ->  xt[c][r][f]  (64B of features contiguous per (r,c))
__global__ void transpose_kernel(const float* __restrict__ x,
                                 float* __restrict__ xt) {
  int idx = blockIdx.x * blockDim.x + threadIdx.x;   // r*C + c, c fastest
  if (idx >= RC) return;
  int c = idx & (C_DIM - 1);
  int r = idx >> 12;
  float v[F_IN];
#pragma unroll
  for (int f = 0; f < F_IN; ++f) v[f] = x[(size_t)f * RC + idx];
  float* o = xt + (((size_t)c * R_DIM + r) << 4);
#pragma unroll
  for (int q = 0; q < 4; ++q)
    ((float4*)o)[q] = make_float4(v[4 * q], v[4 * q + 1], v[4 * q + 2], v[4 * q + 3]);
}

// ---------------------------------------------------------------------------
// Main fused kernel: gather -> GEMM1(WMMA f32) -> ReLU -> GEMM2(WMMA f32)
//                    -> 3x scatter atomic-max
// One wave per 16-row tile of one (set, m) column task.
// ---------------------------------------------------------------------------
template <bool USE_XT>
__global__ __launch_bounds__(256) void mlp_scatter_kernel(
    const float* __restrict__ xsrc,   // xt[c][r][f] if USE_XT, else x[f][r][c]
    const float* __restrict__ w1,     // [64][48]
    const float* __restrict__ b1,     // [64]
    const float* __restrict__ w2,     // [16][64]
    const float* __restrict__ b2,     // [16]
    const int* __restrict__ src,      // [2*3][4096]
    const int* __restrict__ dst,      // [2*3][4096]
    float* __restrict__ out) {        // [16][256][4096]
  __shared__ float lds_a1[8][16][68];          // per-wave a1 staging, padded

  const int lane = threadIdx.x & 31;
  const int wav  = threadIdx.x >> 5;
  const int ln16 = lane & 15;
  const int hi   = lane >> 4;        // 0: lanes 0-15, 1: lanes 16-31
  const int base = hi ? 2 : 0;       // K sub-index pair owned by this half-wave

  const int gwave = blockIdx.x * (blockDim.x >> 5) + wav;
  const int nwave = gridDim.x * (blockDim.x >> 5);

  // ---- preload register-resident B operands (loop invariant) ----
  // GEMM1: B1[kk][t] = (w1T[j0][h], w1T[j0+1][h]) = (w1[h][j0], w1[h][j0+1])
  v2f B1[12][4];
#pragma unroll
  for (int kk = 0; kk < 12; ++kk) {
    const int j0 = 4 * kk + base;
#pragma unroll
    for (int t = 0; t < 4; ++t) {
      const int h = 16 * t + ln16;
      B1[kk][t].x = w1[h * K3F + j0];
      B1[kk][t].y = w1[h * K3F + j0 + 1];
    }
  }
  // GEMM2: B2[kk] = (w2T[j0][o], w2T[j0+1][o]) = (w2[o][j0], w2[o][j0+1])
  v2f B2[16];
#pragma unroll
  for (int kk = 0; kk < 16; ++kk) {
    const int j0 = 4 * kk + base;
    B2[kk].x = w2[ln16 * HID + j0];
    B2[kk].y = w2[ln16 * HID + j0 + 1];
  }
  float bv1[4];
#pragma unroll
  for (int t = 0; t < 4; ++t) bv1[t] = b1[16 * t + ln16];
  const float bv2 = b2[ln16];

  const int ntask = NSETS * L_DIM * (R_DIM / 16);   // 131072

  for (int task = gwave; task < ntask; task += nwave) {   // wave-uniform
    const int rb = task & 15;
    const int m  = (task >> 4) & (L_DIM - 1);
    const int sk = (task >> 16) * 3;                 // s*3
    const int r0 = rb * 16;
    const int r  = r0 + ln16;                        // this lane's A-row

    int scol[3], dcol[3];
#pragma unroll
    for (int k = 0; k < 3; ++k) {
      scol[k] = src[((sk + k) << 12) + m];
      dcol[k] = dst[((sk + k) << 12) + m];
    }

    // ---- GEMM1: a1[16x64] = c[16x48] * w1T, 12 K-steps x 4 N-tiles ----
    v8f acc1[4] = {v8f{}, v8f{}, v8f{}, v8f{}};
#pragma unroll
    for (int kk = 0; kk < 12; ++kk) {
      const int j0 = 4 * kk + base;      // column pair (j0, j0+1) of c
      const int col = scol[j0 >> 4];     // source column in C (same k for pair)
      const int f = j0 & 15;
      v2f a;
      if (USE_XT) {
        const float* p = xsrc + (((size_t)col * R_DIM + r) << 4) + f;
        a.x = p[0];
        a.y = p[1];                      // contiguous, 8B aligned
      } else {
        const size_t bp = (size_t)f * RC + (size_t)r * C_DIM + col;
        a.x = xsrc[bp];
        a.y = xsrc[bp + (size_t)RC];
      }
#pragma unroll
      for (int t = 0; t < 4; ++t)
        acc1[t] = __builtin_amdgcn_wmma_f32_16x16x4_f32(
            false, a, false, B1[kk][t], (short)0, acc1[t], false, false);
    }

    // ---- bias + ReLU, stage a1 into LDS in row-major [r_local][h] ----
#pragma unroll
    for (int t = 0; t < 4; ++t) {
#pragma unroll
      for (int v = 0; v < 8; ++v)
        lds_a1[wav][v + 8 * hi][16 * t + ln16] =
            fmaxf(acc1[t][v] + bv1[t], 0.f);
    }
    // same-wave DS ops are in order (DScnt); no barrier needed (per-wave tile)

    // ---- GEMM2: d[16x16] = a1[16x64] * w2T, 16 K-steps ----
    v8f acc2 = {};
#pragma unroll
    for (int kk = 0; kk < 16; ++kk) {
      const int j0 = 4 * kk + base;
      v2f a;
      a.x = lds_a1[wav][ln16][j0];
      a.y = lds_a1[wav][ln16][j0 + 1];
      acc2 = __builtin_amdgcn_wmma_f32_16x16x4_f32(
          false, a, false, B2[kk], (short)0, acc2, false, false);
    }

    // ---- scatter-max: out[o][r][dcol[k]] = max(., d) for k=0..2 ----
#pragma unroll
    for (int v = 0; v < 8; ++v) {
      const float dval = acc2[v] + bv2;
      const int rr = r0 + v + 8 * hi;
      float* po = out + (size_t)ln16 * RC + (size_t)rr * C_DIM;
#pragma unroll
      for (int k = 0; k < 3; ++k)
        atomicMax(po + dcol[k], dval);   // GLOBAL_ATOMIC_MAX_NUM_F32
    }
  }
}

// ---------------------------------------------------------------------------
extern "C" void kernel_launch(void* const* d_in, const int* in_sizes, int n_in,
                              void* d_out, int out_size, void* d_ws, size_t ws_size,
                              hipStream_t stream) {
  const float* x   = (const float*)d_in[0];
  const float* w1  = (const float*)d_in[1];
  const float* b1  = (const float*)d_in[2];
  const float* w2  = (const float*)d_in[3];
  const float* b2  = (const float*)d_in[4];
  const int*   i0  = (const int*)d_in[5];
  const int*   i1  = (const int*)d_in[6];
  const int*   i2  = (const int*)d_in[7];
  const int*   mix = (const int*)d_in[8];
  float* out = (float*)d_out;

  // workspace layout
  int* inv = (int*)d_ws;                       //  96 KB
  int* src = inv + NSETS * 3 * L_DIM;          //  96 KB
  int* dst = src + NSETS * 3 * L_DIM;          //  96 KB
  const size_t xt_off   = (size_t)1 << 20;
  const size_t xt_bytes = (size_t)RC * F_IN * sizeof(float);  // 64 MB
  float* xt = (float*)((char*)d_ws + xt_off);
  const bool use_xt = ws_size >= xt_off + xt_bytes;

  const int nIdx = NSETS * 3 * L_DIM;
  build_inv_kernel<<<(nIdx + 255) / 256, 256, 0, stream>>>(i0, i1, i2, inv);
  build_srcdst_kernel<<<(nIdx + 255) / 256, 256, 0, stream>>>(i0, i1, i2, mix,
                                                              inv, src, dst);
  const int n4 = (F_OUT * RC) / 4;
  zero_kernel<<<(n4 + 255) / 256, 256, 0, stream>>>((float4*)out, n4);

  if (use_xt) {
    transpose_kernel<<<(RC + 255) / 256, 256, 0, stream>>>(x, xt);
    mlp_scatter_kernel<true><<<1024, 256, 0, stream>>>(xt, w1, b1, w2, b2,
                                                       src, dst, out);
  } else {
    mlp_scatter_kernel<false><<<1024, 256, 0, stream>>>(x, w1, b1, w2, b2,
                                                        src, dst, out);
  }
}